// custom_loss_80934363726057
// MI455X (gfx1250) — compile-verified
//
#include <hip/hip_runtime.h>

typedef float v2f __attribute__((ext_vector_type(2)));
typedef float v8f __attribute__((ext_vector_type(8)));

// normalized 1D gaussian taps for sigma=1.5, ksize=3 (g0, g1, g0), sum = 1
#define G0f 0.30780131f
#define G1f 0.38439748f

// geometry: per-batch image is 512 rows x 1536 flat cols (h*3+c interleaved)
constexpr int RS     = 1536;            // row stride (floats)
constexpr int IMG    = 512 * 1536;      // batch stride (floats)
constexpr int R0 = 5,  R1 = 506;        // valid output rows (inclusive)
constexpr int C0 = 15, C1 = 1520;       // valid output flat cols (inclusive)
constexpr int TM = 16, TN = 128;        // output tile
constexpr int LDSTR = 136;              // LDS row stride (136 % 64 == 8 -> conflict free)
constexpr int NROWT = 32;               // ceil(502/16)
constexpr int NCOLT = 12;               // ceil(1506/128)
constexpr float SCALE = 1.0f / 756012.0f;   // 1/(502*502*3)

__global__ __launch_bounds__(256) void loss_tile_kernel(const float* __restrict__ X,
                                                        const float* __restrict__ Y,
                                                        float* __restrict__ partial)
{
    __shared__ float dT [18 * LDSTR];   // d = x - y, with halo
    __shared__ float muH[18 * LDSTR];   // horizontal filt(d)
    __shared__ float exH[18 * LDSTR];   // horizontal filt(d*d)
    __shared__ float red[8];

    const int tid = threadIdx.x;
    const int r0  = R0 + blockIdx.y * TM;     // first output row of tile
    const int c0  = C0 + blockIdx.x * TN;     // first output flat col of tile
    const int cs0 = c0 - 3;                   // staged col start; (c0-3) % 4 == 0 -> 16B aligned
    const size_t base = (size_t)blockIdx.z * IMG;

    // ---- stage d = x - y into LDS: 18 rows x 34 float4 (136 cols) ----
    for (int u = tid; u < 18 * 34; u += 256) {
        int row  = u / 34, q = u - row * 34;
        int grow = r0 - 1 + row;  if (grow > 511)  grow = 511;   // clamp feeds only discarded outputs
        int gcol = cs0 + 4 * q;   if (gcol > 1532) gcol = 1532;
        const size_t off = base + (size_t)grow * RS + gcol;
        const float4 x4 = *(const float4*)(X + off);
        const float4 y4 = *(const float4*)(Y + off);
        float4 d4;
        d4.x = x4.x - y4.x; d4.y = x4.y - y4.y;
        d4.z = x4.z - y4.z; d4.w = x4.w - y4.w;
        *(float4*)(&dT[row * LDSTR + 4 * q]) = d4;
    }
    __syncthreads();

    // ---- horizontal pass: taps at flat-col stride 3 (channel interleave) ----
    for (int e = tid; e < 18 * 128; e += 256) {
        int row = e >> 7, j = e & 127;
        const float* dr = &dT[row * LDSTR + j];
        float a = dr[0], b = dr[3], c = dr[6];
        muH[row * LDSTR + j] = fmaf(G1f, b,     G0f * (a + c));
        exH[row * LDSTR + j] = fmaf(G1f, b * b, G0f * fmaf(a, a, c * c));
    }
    __syncthreads();

    // ---- vertical pass: OUT(16x16) = G(16x18) * T(18x16), banded G,
    //      K padded 18->20, 5 accumulating v_wmma_f32_16x16x4_f32 chunks ----
    const int wave = tid >> 5;
    const int lane = tid & 31;
    const int m    = lane & 15;     // A: M index / B: N index / D: N index
    const int hi   = lane >> 4;
    const int n0   = wave * 16;     // this wave's 16-column block

    v8f Dmu = {};
    v8f Dex = {};
#pragma unroll
    for (int k = 0; k < 5; ++k) {
        // A[m,p] = g[p-m] for p-m in {0,1,2}; lanes 0-15 hold K={0,1}, 16-31 hold K={2,3}
        int p0 = 4 * k + 2 * hi;
        int t0 = p0 - m, t1 = p0 + 1 - m;
        v2f A;
        A.x = (t0 == 1) ? G1f : ((t0 == 0 || t0 == 2) ? G0f : 0.0f);
        A.y = (t1 == 1) ? G1f : ((t1 == 0 || t1 == 2) ? G0f : 0.0f);
        // B rows: clamp pad rows (>=18) to finite data; their A coeffs are exactly 0
        int pr0 = (p0     > 17) ? 17 : p0;
        int pr1 = (p0 + 1 > 17) ? 17 : p0 + 1;
        v2f Bmu, Bex;
        Bmu.x = muH[pr0 * LDSTR + n0 + m];  Bmu.y = muH[pr1 * LDSTR + n0 + m];
        Bex.x = exH[pr0 * LDSTR + n0 + m];  Bex.y = exH[pr1 * LDSTR + n0 + m];
        Dmu = __builtin_amdgcn_wmma_f32_16x16x4_f32(false, A, false, Bmu, (short)0, Dmu, false, false);
        Dex = __builtin_amdgcn_wmma_f32_16x16x4_f32(false, A, false, Bex, (short)0, Dex, false, false);
    }

    // ---- loss = E[d^2] - E[d]^2, masked by crop bounds, accumulate ----
    float sum = 0.0f;
    const int  outc = c0 + n0 + m;
    const bool cok  = (outc <= C1);
#pragma unroll
    for (int i = 0; i < 8; ++i) {
        int outr = r0 + i + 8 * hi;          // D vgpr i: lanes0-15 M=i, lanes16-31 M=8+i
        float v = Dex[i] - Dmu[i] * Dmu[i];
        if (cok && outr <= R1) sum += v;
    }
#pragma unroll
    for (int off = 16; off >= 1; off >>= 1)
        sum += __shfl_xor(sum, off, 32);
    if (lane == 0) red[wave] = sum;
    __syncthreads();
    if (tid == 0) {
        float t = 0.0f;
#pragma unroll
        for (int w = 0; w < 8; ++w) t += red[w];
        partial[((size_t)blockIdx.z * gridDim.y + blockIdx.y) * gridDim.x + blockIdx.x] = t;
    }
}

// deterministic fixed-order final reduction (graph replays must re-validate d_out)
__global__ __launch_bounds__(256) void reduce_partials(const float* __restrict__ partial,
                                                       float* __restrict__ out, int n)
{
    __shared__ float buf[256];
    float s = 0.0f;
    for (int i = threadIdx.x; i < n; i += 256) s += partial[i];
    buf[threadIdx.x] = s;
    __syncthreads();
    for (int off = 128; off >= 1; off >>= 1) {
        if ((int)threadIdx.x < off) buf[threadIdx.x] += buf[threadIdx.x + off];
        __syncthreads();
    }
    if (threadIdx.x == 0) out[0] = buf[0] * SCALE;
}

extern "C" void kernel_launch(void* const* d_in, const int* in_sizes, int n_in,
                              void* d_out, int out_size, void* d_ws, size_t ws_size,
                              hipStream_t stream)
{
    (void)n_in; (void)out_size; (void)ws_size;
    const float* X = (const float*)d_in[0];
    const float* Y = (const float*)d_in[1];
    const int B = in_sizes[0] / IMG;          // 32
    float* partial = (float*)d_ws;            // NCOLT*NROWT*B floats (48 KB)

    dim3 grid(NCOLT, NROWT, B);
    loss_tile_kernel<<<grid, 256, 0, stream>>>(X, Y, partial);
    reduce_partials<<<1, 256, 0, stream>>>(partial, (float*)d_out, NCOLT * NROWT * B);
}